// EUCClassifier_72129680769356
// MI455X (gfx1250) — compile-verified
//
#include <hip/hip_runtime.h>

// EUC classifier: scores[b,o] = -(||w_o - x_b||^2) / 64
//               = (2 * dot(x_b, w_o) - ||x_b||^2 - ||w_o||^2) / 64
// GEMM on V_WMMA_F32_16X16X4_F32 (full fp32, wave32). Fully unrolled K loop,
// b128 loads feeding two WMMAs each via column-permutation (columns can be
// consumed in any order as long as A and B agree).

typedef __attribute__((ext_vector_type(2))) float v2f;
typedef __attribute__((ext_vector_type(4))) float v4f;
typedef __attribute__((ext_vector_type(8))) float v8f;

#define B_DIM   256
#define OUT_DIM 1000
#define C_DIM   512
#define INV_TEMP (1.0f / 64.0f)

// ---------------------------------------------------------------------------
// Kernel 1: row norms. One wave32 per row; rows 0..255 are x, 256..1255 are w.
// ---------------------------------------------------------------------------
__global__ __launch_bounds__(256) void euc_norms(const float* __restrict__ x,
                                                 const float* __restrict__ w,
                                                 float* __restrict__ norms) {
  const int wave = (int)((blockIdx.x * blockDim.x + threadIdx.x) >> 5);
  const int lane = (int)(threadIdx.x & 31);
  if (wave >= B_DIM + OUT_DIM) return;

  const float* row = (wave < B_DIM) ? (x + (size_t)wave * C_DIM)
                                    : (w + (size_t)(wave - B_DIM) * C_DIM);
  float s = 0.0f;
#pragma unroll
  for (int i = 0; i < C_DIM / (32 * 4); ++i) {      // 4 x float4 per lane
    float4 v = ((const float4*)row)[lane + 32 * i];
    s += v.x * v.x + v.y * v.y + v.z * v.z + v.w * v.w;
  }
#pragma unroll
  for (int off = 16; off > 0; off >>= 1)            // wave32 butterfly reduce
    s += __shfl_xor(s, off, 32);
  if (lane == 0) norms[wave] = s;
}

// ---------------------------------------------------------------------------
// Kernel 2: WMMA GEMM + epilogue.
// Each wave: one 16-row M tile x two 16-col N tiles (2 independent acc chains).
// 512 wave-tasks = 16 mTiles x 32 nGroups; the 8 waves of a block share the
// same 32 weight rows (B reuse in WGP$). Grid = 64 blocks x 256 threads.
//
// Column permutation: chunk s owns columns [8s, 8s+8). Each lane loads a
// float4 at col 8s + 4*hi. WMMA #1 uses elems (0,1): lo lanes supply K0,K1 =
// cols 8s,8s+1; hi lanes supply K2,K3 = cols 8s+4,8s+5. WMMA #2 uses elems
// (2,3): cols {8s+2, 8s+3, 8s+6, 8s+7}. A and B use the identical mapping,
// so the pair of WMMAs sums each column of the chunk exactly once.
// ---------------------------------------------------------------------------
__global__ __launch_bounds__(256) void euc_scores(const float* __restrict__ x,
                                                  const float* __restrict__ w,
                                                  const float* __restrict__ norms,
                                                  float* __restrict__ out) {
  const int lane = (int)(threadIdx.x & 31);
  const int wave = (int)(threadIdx.x >> 5);
  const int l16  = lane & 15;    // A: row within tile / B,C,D: column
  const int hi   = lane >> 4;    // K-half (A/B) / row-half (C,D)

  const int task   = (int)blockIdx.x * 8 + wave;  // 0..511
  const int mTile  = task & 15;                   // 16 M tiles (256 rows)
  const int nGroup = task >> 4;                   // 32 N groups (2 tiles each)

  const int mBase = mTile * 16;
  const int n0 = nGroup * 32 + l16;               // lane's column, tile 0
  const int n1 = n0 + 16;                         // lane's column, tile 1
  // Clamp class index for loads so EXEC stays all-ones through the WMMAs;
  // out-of-range columns are masked at the store.
  const int n0c = (n0 < OUT_DIM) ? n0 : (OUT_DIM - 1);
  const int n1c = (n1 < OUT_DIM) ? n1 : (OUT_DIM - 1);

  // One address computation per stream; all K offsets are instruction
  // immediates (max 8*63*4 = 2016 bytes, fits the 24-bit IOFFSET).
  const float* aPtr  = x + (size_t)(mBase + l16) * C_DIM + 4 * hi;
  const float* b0Ptr = w + (size_t)n0c * C_DIM + 4 * hi;
  const float* b1Ptr = w + (size_t)n1c * C_DIM + 4 * hi;

  v8f acc0 = {};
  v8f acc1 = {};

#pragma unroll
  for (int s = 0; s < C_DIM / 8; ++s) {           // 64 chunks, fully unrolled
    v4f av  = *(const v4f*)(aPtr  + 8 * s);
    v4f b0v = *(const v4f*)(b0Ptr + 8 * s);
    v4f b1v = *(const v4f*)(b1Ptr + 8 * s);

    v2f a01  = __builtin_shufflevector(av,  av,  0, 1);
    v2f a23  = __builtin_shufflevector(av,  av,  2, 3);
    v2f b001 = __builtin_shufflevector(b0v, b0v, 0, 1);
    v2f b023 = __builtin_shufflevector(b0v, b0v, 2, 3);
    v2f b101 = __builtin_shufflevector(b1v, b1v, 0, 1);
    v2f b123 = __builtin_shufflevector(b1v, b1v, 2, 3);

    acc0 = __builtin_amdgcn_wmma_f32_16x16x4_f32(
        false, a01, false, b001, (short)0, acc0, false, false);
    acc1 = __builtin_amdgcn_wmma_f32_16x16x4_f32(
        false, a01, false, b101, (short)0, acc1, false, false);
    acc0 = __builtin_amdgcn_wmma_f32_16x16x4_f32(
        false, a23, false, b023, (short)0, acc0, false, false);
    acc1 = __builtin_amdgcn_wmma_f32_16x16x4_f32(
        false, a23, false, b123, (short)0, acc1, false, false);
  }

  const float ww0 = norms[B_DIM + n0c];
  const float ww1 = norms[B_DIM + n1c];

  // C/D layout: VGPR v, lane -> element [row = v + 8*hi][col = l16].
#pragma unroll
  for (int v = 0; v < 8; ++v) {
    const int m = mBase + v + 8 * hi;
    const float xx = norms[m];
    if (n0 < OUT_DIM)
      out[(size_t)m * OUT_DIM + n0] = (2.0f * acc0[v] - xx - ww0) * INV_TEMP;
    if (n1 < OUT_DIM)
      out[(size_t)m * OUT_DIM + n1] = (2.0f * acc1[v] - xx - ww1) * INV_TEMP;
  }
}

// ---------------------------------------------------------------------------
extern "C" void kernel_launch(void* const* d_in, const int* in_sizes, int n_in,
                              void* d_out, int out_size, void* d_ws, size_t ws_size,
                              hipStream_t stream) {
  (void)in_sizes; (void)n_in; (void)out_size; (void)ws_size;
  const float* x = (const float*)d_in[0];       // (256, 1, 512) f32
  const float* w = (const float*)d_in[1];       // (1000, 512)  f32
  float* out   = (float*)d_out;                 // (256, 1000)  f32
  float* norms = (float*)d_ws;                  // 1256 floats: ||x||^2, ||w||^2

  // 1256 waves -> 157 blocks of 8 waves.
  const int normBlocks = ((B_DIM + OUT_DIM) * 32 + 255) / 256;
  euc_norms<<<normBlocks, 256, 0, stream>>>(x, w, norms);

  // 512 wave-tasks / 8 waves per block = 64 blocks.
  euc_scores<<<64, 256, 0, stream>>>(x, w, norms, out);
}